// HexPlaneField_tt_38732015076050
// MI455X (gfx1250) — compile-verified
//
#include <hip/hip_runtime.h>
#include <math.h>

// ---------------------------------------------------------------------------
// HexPlane field sampling, MI455X (gfx1250, wave32).
//
// d_in[0] pts [N,3] f32 | d_in[1] timestamps [N,1] f32
// d_in[2..19] grids: 3 levels x 6 planes, plane = [32,H,W] f32
// d_in[20] aabb [2,3] f32
// out: [N,96] f32 = per level, elementwise product of 6 bilinear samples.
//
// v3 changes vs v2:
//  * software-pipelined prefetch: addrgen+prefetch of level l+1 is issued
//    before the loads/math of level l, giving every global_prefetch_b8 a
//    full level (~200 VALU cycles/wave) to cover L2 latency into WGP$.
//  * per-lane feature-quad offset folded once into the base pointer.
//  * transpose emits one coalesced b128 store per thread (4 NT strided reads).
// ---------------------------------------------------------------------------

#define NPLANES 18

typedef float v4f __attribute__((ext_vector_type(4)));

// Per-plane tables, index = level*6 + ci. COO = (0,1)(0,2)(0,3)(1,2)(1,3)(2,3)
static constexpr int kW[NPLANES] = {
   64, 64, 64, 64, 64, 64,
  128,128,128,128,128,128,
  256,256,256,256,256,256};
static constexpr int kH[NPLANES] = {
   64, 64, 25, 64, 25, 25,
  128,128, 25,128, 25, 25,
  256,256, 25,256, 25, 25};
static constexpr int kA[6] = {0,0,0,1,1,2};   // coord index on W axis
static constexpr int kB[6] = {1,2,3,2,3,3};   // coord index on H axis
// Float offsets of each transposed plane inside d_ws (plane = H*W*32 floats)
static constexpr int kOFF[NPLANES] = {
        0,  131072,  262144,  313344,  444416,  495616,
   546816, 1071104, 1595392, 1697792, 2222080, 2324480,
  2426880, 4524032, 6621184, 6825984, 8923136, 9127936};
// total = 9,332,736 floats = 35.6 MiB in d_ws

// ---------------------------------------------------------------------------
// Transpose one plane [32, H, W] -> [H, W, 32].
// Thread -> one float4 of 4 consecutive features: 4 strided NT reads (grids
// are read exactly once in this layout), 1 coalesced b128 store.
// ---------------------------------------------------------------------------
__global__ __launch_bounds__(256) void hexplane_transpose(
    const float* __restrict__ src, float* __restrict__ dst, int HW)
{
  int tid = blockIdx.x * blockDim.x + threadIdx.x;   // over HW*8 quads
  if (tid >= HW * 8) return;
  int f4 = tid & 7;        // feature quad
  int hw = tid >> 3;       // h*W + w
  v4f v;
  #pragma unroll
  for (int j = 0; j < 4; ++j)
    v[j] = __builtin_nontemporal_load(src + (f4 * 4 + j) * HW + hw);
  reinterpret_cast<v4f*>(dst)[tid] = v;
}

// ---------------------------------------------------------------------------
// Address generation + prefetch for one level (all dims fold to literals).
// ---------------------------------------------------------------------------
template <int L>
__device__ __forceinline__ void hex_addrgen(
    const float* __restrict__ c, const v4f* __restrict__ Pq,
    int* __restrict__ o00, int* __restrict__ o01,
    int* __restrict__ o10, int* __restrict__ o11,
    float* __restrict__ wxx, float* __restrict__ wyy)
{
  #pragma unroll
  for (int ci = 0; ci < 6; ++ci) {
    const int pi = L * 6 + ci;
    const int W  = kW[pi], H = kH[pi];
    const float cx = c[kA[ci]], cy = c[kB[ci]];

    // align_corners=True, padding_mode='border'
    float x = fminf(fmaxf((cx + 1.0f) * 0.5f * (float)(W - 1), 0.0f), (float)(W - 1));
    float y = fminf(fmaxf((cy + 1.0f) * 0.5f * (float)(H - 1), 0.0f), (float)(H - 1));
    int x0 = (int)x;                 // x >= 0 -> trunc == floor
    int y0 = (int)y;
    int x1 = min(x0 + 1, W - 1);
    int y1 = min(y0 + 1, H - 1);
    wxx[ci] = x - (float)x0;
    wyy[ci] = y - (float)y0;

    const int base = kOFF[pi] >> 2;            // float4 units (q is in Pq)
    o00[ci] = base + (y0 * W + x0) * 8;
    o01[ci] = base + (y0 * W + x1) * 8;
    o10[ci] = base + (y1 * W + x0) * 8;
    o11[ci] = base + (y1 * W + x1) * 8;

    __builtin_prefetch(Pq + o00[ci], 0, 3);
    __builtin_prefetch(Pq + o01[ci], 0, 3);
    __builtin_prefetch(Pq + o10[ci], 0, 3);
    __builtin_prefetch(Pq + o11[ci], 0, 3);
  }
}

// ---------------------------------------------------------------------------
// Load + bilinear + 6-way product for one level (vector <4 x float> math ->
// V_PK_FMA_F32 / VOPD dual-issue candidates).
// ---------------------------------------------------------------------------
__device__ __forceinline__ v4f hex_level(
    const v4f* __restrict__ Pq,
    const int* __restrict__ o00, const int* __restrict__ o01,
    const int* __restrict__ o10, const int* __restrict__ o11,
    const float* __restrict__ wxx, const float* __restrict__ wyy)
{
  v4f prod = {1.0f, 1.0f, 1.0f, 1.0f};
  #pragma unroll
  for (int ci = 0; ci < 6; ++ci) {
    v4f v00 = Pq[o00[ci]];
    v4f v01 = Pq[o01[ci]];
    v4f v10 = Pq[o10[ci]];
    v4f v11 = Pq[o11[ci]];
    const float wx = wxx[ci], wy = wyy[ci];
    const float ox = 1.0f - wx, oy = 1.0f - wy;
    // exact reference form: (v00*ox + v01*wx)*oy + (v10*ox + v11*wx)*wy
    v4f top = v00 * ox + v01 * wx;
    v4f bot = v10 * ox + v11 * wx;
    prod *= top * oy + bot * wy;
  }
  return prod;
}

// ---------------------------------------------------------------------------
// Main kernel: thread t -> point p = t>>3, feature-quad q = t&7.
// 8 lanes of a point cover one 128B feature block per corner (coalesced b128).
// Levels are software-pipelined: prefetch(l+1) issued before loads/math(l).
// ---------------------------------------------------------------------------
__global__ __launch_bounds__(256) void hexplane_sample(
    const float* __restrict__ pts,
    const float* __restrict__ ts,
    const float* __restrict__ aabb,
    const float* __restrict__ planes,   // d_ws, transposed planes
    float*       __restrict__ out,
    int N)
{
  int t = blockIdx.x * blockDim.x + threadIdx.x;
  int p = t >> 3;
  int q = t & 7;
  if (p >= N) return;

  // p_norm = (pts - aabb[0]) * (2 / (aabb[1]-aabb[0])) - 1 ; time stays raw
  float c[4];
  #pragma unroll
  for (int i = 0; i < 3; ++i) {
    float a0 = aabb[i], a1 = aabb[3 + i];
    c[i] = (pts[p * 3 + i] - a0) * (2.0f / (a1 - a0)) - 1.0f;
  }
  c[3] = ts[p];

  const v4f* __restrict__ Pq = reinterpret_cast<const v4f*>(planes) + q;
  v4f* __restrict__ out4 = reinterpret_cast<v4f*>(out);
  const int ob = p * 24 + q;   // out[p, l*32 + 4q..] -> float4 idx ob + l*8

  // Double-buffered per-level state (kept in registers; all indices literal).
  int   oA00[6], oA01[6], oA10[6], oA11[6], oB00[6], oB01[6], oB10[6], oB11[6];
  float wAx[6], wAy[6], wBx[6], wBy[6];

  hex_addrgen<0>(c, Pq, oA00, oA01, oA10, oA11, wAx, wAy);
  hex_addrgen<1>(c, Pq, oB00, oB01, oB10, oB11, wBx, wBy);

  v4f r0 = hex_level(Pq, oA00, oA01, oA10, oA11, wAx, wAy);
  __builtin_nontemporal_store(r0, out4 + ob);          // NT: write-once stream

  hex_addrgen<2>(c, Pq, oA00, oA01, oA10, oA11, wAx, wAy);

  v4f r1 = hex_level(Pq, oB00, oB01, oB10, oB11, wBx, wBy);
  __builtin_nontemporal_store(r1, out4 + ob + 8);

  v4f r2 = hex_level(Pq, oA00, oA01, oA10, oA11, wAx, wAy);
  __builtin_nontemporal_store(r2, out4 + ob + 16);
}

// ---------------------------------------------------------------------------
extern "C" void kernel_launch(void* const* d_in, const int* in_sizes, int n_in,
                              void* d_out, int out_size, void* d_ws, size_t ws_size,
                              hipStream_t stream) {
  (void)n_in; (void)out_size; (void)ws_size;

  const float* pts  = (const float*)d_in[0];
  const float* ts   = (const float*)d_in[1];
  const float* aabb = (const float*)d_in[20];
  float*       ws   = (float*)d_ws;
  float*       out  = (float*)d_out;

  const int N = in_sizes[0] / 3;

  // Stage 1: transpose all 18 planes [32,H,W] -> [H,W,32] into workspace.
  for (int pi = 0; pi < NPLANES; ++pi) {
    const float* src = (const float*)d_in[2 + pi];
    const int HW = kH[pi] * kW[pi];
    const int quads = HW * 8;
    hexplane_transpose<<<(quads + 255) / 256, 256, 0, stream>>>(src, ws + kOFF[pi], HW);
  }

  // Stage 2: sample. 8 threads per point (feature quads), wave32-friendly.
  const long threads = (long)N * 8;
  const int  blocks  = (int)((threads + 255) / 256);
  hexplane_sample<<<blocks, 256, 0, stream>>>(pts, ts, aabb, ws, out, N);
}